// GAT_14577119003075
// MI455X (gfx1250) — compile-verified
//
#include <hip/hip_runtime.h>
#include <math.h>

// ---------------- types for WMMA ----------------
typedef float v2f __attribute__((ext_vector_type(2)));
typedef float v8f __attribute__((ext_vector_type(8)));

#ifndef NEG_SLOPE
#define NEG_SLOPE 0.2f
#endif

// =====================================================================
// GEMM: H[M,NOUT] = A[M,128] @ W[128,NOUT]   (fp32, V_WMMA_F32_16X16X4_F32)
// One 16x16 output tile per wave32; K-loop of 32 chained WMMAs.
// NOUT is a compile-time constant so all inner-loop addresses are
// immediate offsets from fixed base pointers (no per-iteration mul64).
// =====================================================================
template <int NOUT>
__global__ void gat_gemm_wmma_f32(const float* __restrict__ A,
                                  const float* __restrict__ W,
                                  float* __restrict__ H,
                                  int M) {
  constexpr int NTN = NOUT / 16;                   // N tiles (8 or 4)
  const int lane = (int)(threadIdx.x & 31u);
  const int wave = (int)(threadIdx.x >> 5);
  const int wavesPerBlock = (int)(blockDim.x >> 5);
  const int tile = (int)blockIdx.x * wavesPerBlock + wave;
  const int mt = tile / NTN;
  const int nt = tile - mt * NTN;
  if (mt * 16 >= M) return;                        // wave-uniform: EXEC stays full

  const int mrow = mt * 16 + (lane & 15);          // A row this lane supplies
  const int mr   = (mrow < M) ? mrow : 0;          // clamp (M is 16-aligned here)
  const int koff = (lane >> 4) * 2;                // K sub-offset per A/B layout
  const int ncol = nt * 16 + (lane & 15);          // B/D column this lane supplies

  // Fixed base pointers; all K-loop offsets are compile-time immediates.
  const float* __restrict__ aptr  = A + (size_t)mr * 128 + koff;
  const float* __restrict__ bptr0 = W + (size_t)koff * NOUT + ncol;       // row k0+koff
  const float* __restrict__ bptr1 = bptr0 + NOUT;                         // row k0+koff+1

  v8f c = {};
  #pragma unroll
  for (int k0 = 0; k0 < 128; k0 += 4) {
    const float2 av = *reinterpret_cast<const float2*>(aptr + k0); // 8B, imm offset
    v2f a; a.x = av.x; a.y = av.y;
    v2f b;
    b.x = bptr0[(size_t)k0 * NOUT];                                // imm offset
    b.y = bptr1[(size_t)k0 * NOUT];                                // imm offset
    c = __builtin_amdgcn_wmma_f32_16x16x4_f32(
            /*neg_a=*/false, a, /*neg_b=*/false, b,
            /*c_mod=*/(short)0, c, /*reuse_a=*/false, /*reuse_b=*/false);
  }

  // D layout: VGPR j -> M = j + 8*(lane>=16), N = lane&15
  const int mbase = mt * 16 + (lane >> 4) * 8;
  float* __restrict__ dptr = H + (size_t)mbase * NOUT + ncol;
  if (mbase + 7 < M) {
    #pragma unroll
    for (int j = 0; j < 8; ++j) dptr[(size_t)j * NOUT] = c[j];     // imm offsets
  } else {
    #pragma unroll
    for (int j = 0; j < 8; ++j)
      if (mbase + j < M) dptr[(size_t)j * NOUT] = c[j];
  }
}

// =====================================================================
// Attention logits: al[n,h] = <H[n,h,:], a_src[h,:]>, ar likewise (C=64)
// i = n*heads + h  ->  H row offset is exactly i*64.
// =====================================================================
__global__ void gat_attn_dots(const float* __restrict__ H,
                              const float* __restrict__ a_src,
                              const float* __restrict__ a_dst,
                              float* __restrict__ al, float* __restrict__ ar,
                              int Nn, int heads) {
  const int i = (int)(blockIdx.x * blockDim.x + threadIdx.x);
  if (i >= Nn * heads) return;
  const int h = i % heads;
  const float4* row = reinterpret_cast<const float4*>(H + (size_t)i * 64);
  const float4* as  = reinterpret_cast<const float4*>(a_src + (size_t)h * 64);
  const float4* ad  = reinterpret_cast<const float4*>(a_dst + (size_t)h * 64);
  float s1 = 0.f, s2 = 0.f;
  #pragma unroll
  for (int c = 0; c < 16; ++c) {
    const float4 v = row[c], a = as[c], d = ad[c];
    s1 += v.x * a.x + v.y * a.y + v.z * a.z + v.w * a.w;
    s2 += v.x * d.x + v.y * d.y + v.z * d.z + v.w * d.w;
  }
  al[i] = s1;
  ar[i] = s2;
}

// =====================================================================
// Fill helper
// =====================================================================
__global__ void gat_fill_f32(float* __restrict__ p, float v, long long n) {
  const long long i = (long long)blockIdx.x * blockDim.x + threadIdx.x;
  if (i < n) p[i] = v;
}

__device__ inline void atomicMaxF(float* addr, float val) {
  unsigned int* ua = reinterpret_cast<unsigned int*>(addr);
  unsigned int old = *ua;
  while (true) {
    const float cur = __uint_as_float(old);
    if (cur >= val) break;
    const unsigned int assumed = old;
    old = atomicCAS(ua, assumed, __float_as_uint(val));
    if (old == assumed) break;
  }
}

__device__ inline void edge_endpoints(const int* __restrict__ src,
                                      const int* __restrict__ dst,
                                      int e, int E_, int& s, int& d) {
  if (e < E_) { s = src[e]; d = dst[e]; }
  else        { s = e - E_; d = e - E_; }   // self loops
}

// =====================================================================
// Pass A: raw score = LeakyReLU(al[src]+ar[dst]); segment-max into mx[dst]
// =====================================================================
__global__ void gat_edge_score_max(const int* __restrict__ src,
                                   const int* __restrict__ dst,
                                   const float* __restrict__ al,
                                   const float* __restrict__ ar,
                                   float* __restrict__ esc,
                                   float* __restrict__ mx,
                                   int E_, int Nn, int heads) {
  const long long i = (long long)blockIdx.x * blockDim.x + threadIdx.x;
  const long long tot = (long long)(E_ + Nn) * heads;
  if (i >= tot) return;
  const int e = (int)(i / heads);
  const int h = (int)(i - (long long)e * heads);
  int s, d; edge_endpoints(src, dst, e, E_, s, d);
  float sc = al[(size_t)s * heads + h] + ar[(size_t)d * heads + h];
  sc = (sc > 0.f) ? sc : NEG_SLOPE * sc;
  esc[i] = sc;
  atomicMaxF(&mx[(size_t)d * heads + h], sc);
}

// =====================================================================
// Pass B: esc = exp(esc - mx[dst]); segment-sum into ssum[dst]
// =====================================================================
__global__ void gat_edge_exp_sum(const int* __restrict__ src,
                                 const int* __restrict__ dst,
                                 float* __restrict__ esc,
                                 const float* __restrict__ mx,
                                 float* __restrict__ ssum,
                                 int E_, int Nn, int heads) {
  const long long i = (long long)blockIdx.x * blockDim.x + threadIdx.x;
  const long long tot = (long long)(E_ + Nn) * heads;
  if (i >= tot) return;
  const int e = (int)(i / heads);
  const int h = (int)(i - (long long)e * heads);
  int s, d; edge_endpoints(src, dst, e, E_, s, d);
  const float v = expf(esc[i] - mx[(size_t)d * heads + h]);
  esc[i] = v;
  atomicAdd(&ssum[(size_t)d * heads + h], v);
}

// =====================================================================
// Pass B2: esc = esc / (ssum[dst] + 1e-16)
// =====================================================================
__global__ void gat_edge_norm(const int* __restrict__ src,
                              const int* __restrict__ dst,
                              float* __restrict__ esc,
                              const float* __restrict__ ssum,
                              int E_, int Nn, int heads) {
  const long long i = (long long)blockIdx.x * blockDim.x + threadIdx.x;
  const long long tot = (long long)(E_ + Nn) * heads;
  if (i >= tot) return;
  const int e = (int)(i / heads);
  const int h = (int)(i - (long long)e * heads);
  int s, d; edge_endpoints(src, dst, e, E_, s, d);
  (void)s;
  esc[i] = esc[i] / (ssum[(size_t)d * heads + h] + 1e-16f);
}

// =====================================================================
// Pass C: acc[dst, c] += alpha[e, c/64] * H[src, c]
// one thread per (edge, channel)
// =====================================================================
__global__ void gat_edge_aggregate(const int* __restrict__ src,
                                   const int* __restrict__ dst,
                                   const float* __restrict__ esc,
                                   const float* __restrict__ H,
                                   float* __restrict__ acc,
                                   int E_, int Nn, int heads) {
  const int cht = heads * 64;
  const long long i = (long long)blockIdx.x * blockDim.x + threadIdx.x;
  const long long tot = (long long)(E_ + Nn) * cht;
  if (i >= tot) return;
  const int e = (int)(i / cht);
  const int c = (int)(i - (long long)e * cht);
  int s, d; edge_endpoints(src, dst, e, E_, s, d);
  const float alpha = esc[(size_t)e * heads + (c >> 6)];
  atomicAdd(&acc[(size_t)d * cht + c], H[(size_t)s * cht + c] * alpha);
}

// =====================================================================
// Epilogue: out = acc + bias, optional ELU (in-place safe)
// =====================================================================
__global__ void gat_bias_act(const float* __restrict__ acc,
                             const float* __restrict__ bias,
                             float* __restrict__ out,
                             int Nn, int cht, int do_elu) {
  const long long i = (long long)blockIdx.x * blockDim.x + threadIdx.x;
  const long long tot = (long long)Nn * cht;
  if (i >= tot) return;
  float v = acc[i] + bias[(int)(i % cht)];
  if (do_elu) v = (v > 0.f) ? v : (expf(v) - 1.f);
  out[i] = v;
}

// =====================================================================
// Host orchestration
// =====================================================================
static inline long long cdiv_ll(long long a, long long b) { return (a + b - 1) / b; }

extern "C" void kernel_launch(void* const* d_in, const int* in_sizes, int n_in,
                              void* d_out, int out_size, void* d_ws, size_t ws_size,
                              hipStream_t stream) {
  const float* x      = (const float*)d_in[0];
  const int*   ei     = (const int*)  d_in[1];
  const float* W1     = (const float*)d_in[2];
  const float* a_src1 = (const float*)d_in[3];
  const float* a_dst1 = (const float*)d_in[4];
  const float* b1     = (const float*)d_in[5];
  const float* W2     = (const float*)d_in[6];
  const float* a_src2 = (const float*)d_in[7];
  const float* a_dst2 = (const float*)d_in[8];
  const float* b2     = (const float*)d_in[9];
  const float* W3     = (const float*)d_in[10];
  const float* a_src3 = (const float*)d_in[11];
  const float* a_dst3 = (const float*)d_in[12];
  const float* b3     = (const float*)d_in[13];

  const int Nn = in_sizes[0] / 128;     // 50000
  const int E_ = in_sizes[1] / 2;       // 800000
  const int Etot = E_ + Nn;
  const int* src = ei;
  const int* dst = ei + E_;

  // workspace layout (floats)
  float* ws = (float*)d_ws;
  size_t off = 0;
  float* buf0 = ws + off; off += (size_t)Nn * 128;   // GEMM output H
  float* buf1 = ws + off; off += (size_t)Nn * 128;   // acc / next-layer features
  float* al   = ws + off; off += (size_t)Nn * 2;
  float* ar   = ws + off; off += (size_t)Nn * 2;
  float* mx   = ws + off; off += (size_t)Nn * 2;
  float* ssum = ws + off; off += (size_t)Nn * 2;
  float* esc  = ws + off; off += (size_t)Etot * 2;
  (void)ws_size;

  const int TB = 256;

  auto layer = [&](const float* feat, const float* W, const float* asrc,
                   const float* adst, const float* bias,
                   float* hbuf, float* accbuf, float* outbuf,
                   int heads, int do_elu) {
    const int cht = heads * 64;
    // 1) H = feat @ W (WMMA f32), NOUT templated for immediate addressing
    {
      const long long tiles = cdiv_ll(Nn, 16) * (cht / 16);
      const long long blocks = cdiv_ll(tiles, TB / 32);
      if (cht == 128)
        gat_gemm_wmma_f32<128><<<(unsigned)blocks, TB, 0, stream>>>(feat, W, hbuf, Nn);
      else
        gat_gemm_wmma_f32<64><<<(unsigned)blocks, TB, 0, stream>>>(feat, W, hbuf, Nn);
    }
    // 2) attention logits
    {
      const long long t = (long long)Nn * heads;
      gat_attn_dots<<<(unsigned)cdiv_ll(t, TB), TB, 0, stream>>>(hbuf, asrc, adst,
                                                                 al, ar, Nn, heads);
    }
    // 3) init reductions + acc
    {
      const long long tnh = (long long)Nn * heads;
      gat_fill_f32<<<(unsigned)cdiv_ll(tnh, TB), TB, 0, stream>>>(mx,  -1e30f, tnh);
      gat_fill_f32<<<(unsigned)cdiv_ll(tnh, TB), TB, 0, stream>>>(ssum, 0.f,   tnh);
      const long long tacc = (long long)Nn * cht;
      gat_fill_f32<<<(unsigned)cdiv_ll(tacc, TB), TB, 0, stream>>>(accbuf, 0.f, tacc);
    }
    // 4) edge passes
    {
      const long long te = (long long)Etot * heads;
      gat_edge_score_max<<<(unsigned)cdiv_ll(te, TB), TB, 0, stream>>>(
          src, dst, al, ar, esc, mx, E_, Nn, heads);
      gat_edge_exp_sum<<<(unsigned)cdiv_ll(te, TB), TB, 0, stream>>>(
          src, dst, esc, mx, ssum, E_, Nn, heads);
      gat_edge_norm<<<(unsigned)cdiv_ll(te, TB), TB, 0, stream>>>(
          src, dst, esc, ssum, E_, Nn, heads);
      const long long tagg = (long long)Etot * cht;
      gat_edge_aggregate<<<(unsigned)cdiv_ll(tagg, TB), TB, 0, stream>>>(
          src, dst, esc, hbuf, accbuf, E_, Nn, heads);
    }
    // 5) bias + activation
    {
      const long long t = (long long)Nn * cht;
      gat_bias_act<<<(unsigned)cdiv_ll(t, TB), TB, 0, stream>>>(
          accbuf, bias, outbuf, Nn, cht, do_elu);
    }
  };

  // Layer 1: x -> buf1 (ELU)
  layer(x,    W1, a_src1, a_dst1, b1, buf0, buf1, buf1, /*heads=*/2, /*elu=*/1);
  // Layer 2: buf1 -> buf1 (ELU); GEMM reads buf1 into buf0 before acc reuses buf1
  layer(buf1, W2, a_src2, a_dst2, b2, buf0, buf1, buf1, /*heads=*/2, /*elu=*/1);
  // Layer 3: buf1 -> d_out (heads=1, no ELU; mean over 1 head == identity)
  layer(buf1, W3, a_src3, a_dst3, b3, buf0, (float*)d_out, (float*)d_out,
        /*heads=*/1, /*elu=*/0);
}